// LIFCell_42631845380788
// MI455X (gfx1250) — compile-verified
//
#include <hip/hip_runtime.h>

// LIF recurrence: v += (x - v)/TAU; s = sigmoid(v - TH); v *= (1 - s); out = s
// Shapes: x [B=32, T=512, D=2048] f32, out same. Sequential in T, parallel in B*D.

#define TAU_F 2.0f
#define TH_F  1.0f

typedef float v2f __attribute__((ext_vector_type(2)));

constexpr int B = 32;
constexpr int T = 512;
constexpr int D = 2048;
constexpr int VEC = 2;                   // channels per thread (float2)
constexpr int U   = 8;                   // timestep lookahead (register double buffer)
constexpr int NTHREADS = B * D / VEC;    // 32768
constexpr int BLOCK = 64;                // 2 wave32s per workgroup -> 512 WGs

__global__ __launch_bounds__(BLOCK, 1)
void lif_stream_kernel(const float* __restrict__ x, float* __restrict__ out) {
    const int tid = blockIdx.x * BLOCK + threadIdx.x;
    if (tid >= NTHREADS) return;

    const int b  = tid / (D / VEC);
    const int dp = tid % (D / VEC);
    const size_t base = (size_t)b * T * D + (size_t)dp * VEC;
    const float* __restrict__ xp = x + base;
    float* __restrict__ sp = out + base;

    float v0 = 0.0f, v1 = 0.0f;
    v2f cur[U];
    v2f nxt[U] = {};

    // Preload first chunk of U timesteps (independent loads, deep in flight).
#pragma unroll
    for (int i = 0; i < U; ++i)
        cur[i] = *(const v2f*)(xp + (size_t)i * D);

    for (int t = 0; t < T; t += U) {
        // Far-ahead prefetch into cache hierarchy: gfx1250 global_prefetch_b8.
        if (t + 4 * U < T) {
#pragma unroll
            for (int i = 0; i < U; ++i)
                __builtin_prefetch(xp + (size_t)(t + 4 * U + i) * D, 0, 1);
        }
        // Issue next chunk's loads before consuming the current chunk.
        if (t + U < T) {
#pragma unroll
            for (int i = 0; i < U; ++i)
                nxt[i] = *(const v2f*)(xp + (size_t)(t + U + i) * D);
        }
        // Consume current chunk: two independent LIF chains per thread.
#pragma unroll
        for (int i = 0; i < U; ++i) {
            const v2f xv = cur[i];
            v0 += (xv.x - v0) * (1.0f / TAU_F);
            v1 += (xv.y - v1) * (1.0f / TAU_F);
            const float s0 = 1.0f / (1.0f + __expf(TH_F - v0));
            const float s1 = 1.0f / (1.0f + __expf(TH_F - v1));
            v0 *= (1.0f - s0);
            v1 *= (1.0f - s1);
            v2f sv;
            sv.x = s0;
            sv.y = s1;
            // NT store: rinse output through L2 so x (128MiB) stays resident
            // in the 192MB L2 across graph replays.
            __builtin_nontemporal_store(sv, (v2f*)(sp + (size_t)(t + i) * D));
        }
#pragma unroll
        for (int i = 0; i < U; ++i) cur[i] = nxt[i];
    }
}

extern "C" void kernel_launch(void* const* d_in, const int* in_sizes, int n_in,
                              void* d_out, int out_size, void* d_ws, size_t ws_size,
                              hipStream_t stream) {
    (void)in_sizes; (void)n_in; (void)out_size; (void)d_ws; (void)ws_size;
    const float* x = (const float*)d_in[0];
    float* out = (float*)d_out;
    const int grid = NTHREADS / BLOCK; // 512 workgroups
    lif_stream_kernel<<<grid, BLOCK, 0, stream>>>(x, out);
}